// QuantumLayer_26465588478332
// MI455X (gfx1250) — compile-verified
//
#include <hip/hip_runtime.h>
#include <stdint.h>

// ---------------- problem constants ----------------
#define MM      12          // modes
#define NPH     5           // photons
#define NTHETA  12
#define NXF     8
#define NPAR    20
#define OUTF    10
#define SSTATES 4368        // C(16,5)
#define NMASK   32          // 2^NPH
#define TWO_PI_F 6.28318530717958647692f
#define RSQRT2_F 0.70710678118654752440f

typedef float v2f __attribute__((ext_vector_type(2)));
typedef float v8f __attribute__((ext_vector_type(8)));

// ---------------------------------------------------------------------------
// Init kernel: unrank combinations_with_replacement(12, 5) (lexicographic via
// strictly-increasing combos of 16 choose 5), pack the 5 indices into a u32,
// compute 1/NORM (product of factorials of multiplicities), and optionally
// pad W (S x 10) to Wp (S x 16) for the branchless WMMA B operand.
// ---------------------------------------------------------------------------
__global__ __launch_bounds__(256)
void qinit(uint32_t* __restrict__ idxPack, float* __restrict__ invnorm,
           const float* __restrict__ W, float* __restrict__ Wp)
{
    const int s = blockIdx.x * blockDim.x + threadIdx.x;
    if (s >= SSTATES) return;

    // binomials C(n,k), n in [0,15], k in [0,4]
    int bin[16][5];
    for (int n = 0; n < 16; ++n) {
        bin[n][0] = 1;
        for (int k = 1; k < 5; ++k)
            bin[n][k] = (n == 0) ? 0 : (bin[n-1][k-1] + bin[n-1][k]);
    }

    int rank = s;
    int v = 0;
    int r[5];
    for (int i = 0; i < 5; ++i) {
        for (;;) {
            const int cnt = bin[15 - v][4 - i];   // combos with a_i == v
            if (rank < cnt) { r[i] = v - i; ++v; break; }
            rank -= cnt; ++v;
        }
    }

    uint32_t pk = 0;
    for (int i = 0; i < 5; ++i) pk |= ((uint32_t)r[i]) << (4 * i);

    const int fact[6] = {1, 1, 2, 6, 24, 120};
    int norm = 1, run = 1;
    for (int i = 1; i < 5; ++i) {
        if (r[i] == r[i-1]) ++run;
        else { norm *= fact[run]; run = 1; }
    }
    norm *= fact[run];

    idxPack[s] = pk;
    invnorm[s] = 1.0f / (float)norm;

    if (Wp) {
        for (int o = 0; o < 16; ++o)
            Wp[(size_t)s * 16 + o] = (o < OUTF) ? W[(size_t)s * OUTF + o] : 0.0f;
    }
}

// ---------------------------------------------------------------------------
// Main kernel: one block per batch element.
//   phase 1: threads 0..4 evolve column c of U (starts as e_c) through the 20
//            phase + beamsplitter layers in registers, write U[:, :5] to LDS.
//   phase 2: all threads compute rs[mask][row] = sum of selected columns.
//   phase 3: strided state loop: Ryser accumulation over 31 nonzero masks,
//            prob = |perm|^2 * invnorm.
//   FUSED:   accumulate prob * W row into 10 partials, deterministic
//            shuffle + LDS tree reduce, write out.
//   !FUSED:  write prob row to global for the WMMA GEMM.
// ---------------------------------------------------------------------------
template <bool FUSED>
__global__ __launch_bounds__(256)
void qmain(const float* __restrict__ x, const float* __restrict__ thetas,
           const uint32_t* __restrict__ idxPack, const float* __restrict__ invnorm,
           float* __restrict__ probsOut,
           const float* __restrict__ W, const float* __restrict__ bias,
           float* __restrict__ out)
{
    __shared__ float sUr[MM * NPH];
    __shared__ float sUi[MM * NPH];
    __shared__ float sRs[NMASK * MM * 2];    // interleaved re,im -> b64 LDS loads
    __shared__ float sRed[8][OUTF];

    const int b   = blockIdx.x;
    const int tid = threadIdx.x;

    // ---- phase 1: build the 5 needed columns of U ----
    if (tid < NPH) {
        float ur[MM], ui[MM];
        #pragma unroll
        for (int rr = 0; rr < MM; ++rr) { ur[rr] = (rr == tid) ? 1.0f : 0.0f; ui[rr] = 0.0f; }

        #pragma unroll
        for (int k = 0; k < NPAR; ++k) {
            const int mode = k % MM;
            const float phi = (k < NTHETA) ? thetas[k]
                                           : x[b * NXF + (k - NTHETA)] * TWO_PI_F;
            float sn, cs;
            __sincosf(phi, &sn, &cs);
            {   // multiply row `mode` by e^{i phi}
                const float tr = ur[mode] * cs - ui[mode] * sn;
                const float ti = ur[mode] * sn + ui[mode] * cs;
                ur[mode] = tr; ui[mode] = ti;
            }
            // beamsplitter layer: new_i = r*u_i + i*r*u_j ; new_j = i*r*u_i + r*u_j
            const int st = k & 1;
            #pragma unroll
            for (int i = st; i + 1 < MM - st; i += 2) {
                const float air = ur[i],   aii = ui[i];
                const float ajr = ur[i+1], aji = ui[i+1];
                ur[i]   = RSQRT2_F * (air - aji);
                ui[i]   = RSQRT2_F * (aii + ajr);
                ur[i+1] = RSQRT2_F * (ajr - aii);
                ui[i+1] = RSQRT2_F * (aji + air);
            }
        }
        #pragma unroll
        for (int rr = 0; rr < MM; ++rr) {
            sUr[rr * NPH + tid] = ur[rr];
            sUi[rr * NPH + tid] = ui[rr];
        }
    }
    __syncthreads();

    // ---- phase 2: rs[mask][row] = sum over selected columns ----
    for (int t = tid; t < NMASK * MM; t += 256) {
        const int mask = t / MM;
        const int row  = t % MM;
        float re = 0.0f, im = 0.0f;
        #pragma unroll
        for (int j = 0; j < NPH; ++j) {
            if ((mask >> j) & 1) {
                re += sUr[row * NPH + j];
                im += sUi[row * NPH + j];
            }
        }
        sRs[2 * (mask * MM + row) + 0] = re;
        sRs[2 * (mask * MM + row) + 1] = im;
    }
    __syncthreads();

    // ---- phase 3: Ryser permanent per state ----
    float part[OUTF] = {};

    for (int s = tid; s < SSTATES; s += 256) {
        const uint32_t pk = idxPack[s];
        const int i0 =  pk        & 15;
        const int i1 = (pk >> 4)  & 15;
        const int i2 = (pk >> 8)  & 15;
        const int i3 = (pk >> 12) & 15;
        const int i4 = (pk >> 16) & 15;

        float ar = 0.0f, ai = 0.0f;
        #pragma unroll
        for (int m = 1; m < NMASK; ++m) {      // mask 0 contributes exactly 0
            const float sgn = (__popc(m) & 1) ? 1.0f : -1.0f;  // (-1)^(popc+5)
            const int base = 2 * (m * MM);
            float pr = sRs[base + 2*i0 + 0];
            float pi = sRs[base + 2*i0 + 1];
            float qr, qi, t_;
            qr = sRs[base + 2*i1 + 0]; qi = sRs[base + 2*i1 + 1];
            t_ = pr*qr - pi*qi; pi = pr*qi + pi*qr; pr = t_;
            qr = sRs[base + 2*i2 + 0]; qi = sRs[base + 2*i2 + 1];
            t_ = pr*qr - pi*qi; pi = pr*qi + pi*qr; pr = t_;
            qr = sRs[base + 2*i3 + 0]; qi = sRs[base + 2*i3 + 1];
            t_ = pr*qr - pi*qi; pi = pr*qi + pi*qr; pr = t_;
            qr = sRs[base + 2*i4 + 0]; qi = sRs[base + 2*i4 + 1];
            t_ = pr*qr - pi*qi; pi = pr*qi + pi*qr; pr = t_;
            ar += sgn * pr; ai += sgn * pi;
        }
        const float prob = (ar * ar + ai * ai) * invnorm[s];

        if (FUSED) {
            #pragma unroll
            for (int o = 0; o < OUTF; ++o)
                part[o] += prob * W[(size_t)s * OUTF + o];
        } else {
            probsOut[(size_t)b * SSTATES + s] = prob;
        }
    }

    if (FUSED) {
        // deterministic reduction: intra-wave shuffles, then LDS across 8 waves
        #pragma unroll
        for (int o = 0; o < OUTF; ++o) {
            float v = part[o];
            v += __shfl_xor(v, 16, 32);
            v += __shfl_xor(v,  8, 32);
            v += __shfl_xor(v,  4, 32);
            v += __shfl_xor(v,  2, 32);
            v += __shfl_xor(v,  1, 32);
            part[o] = v;
        }
        const int wv = tid >> 5, ln = tid & 31;
        if (ln == 0) {
            #pragma unroll
            for (int o = 0; o < OUTF; ++o) sRed[wv][o] = part[o];
        }
        __syncthreads();
        if (tid < OUTF) {
            float acc = 0.0f;
            #pragma unroll
            for (int w = 0; w < 8; ++w) acc += sRed[w][tid];
            out[(size_t)b * OUTF + tid] = acc + bias[tid];
        }
    }
}

// ---------------------------------------------------------------------------
// WMMA GEMM: out(nb x 10) = probs(nb x S) @ Wp(S x 16)[: , :10] + bias
// One wave per 16-row M tile, v_wmma_f32_16x16x4_f32, K loop of S/4 = 1092.
// A 16x4 layout: lanes 0-15 M=lane K={0,1}; lanes 16-31 M=lane-16 K={2,3}.
// B 4x16 layout: lanes 0-15 N=lane K={0,1}; lanes 16-31 N=lane-16 K={2,3}.
// D layout: VGPR i -> M = (lane<16 ? i : 8+i), N = lane&15.
// ---------------------------------------------------------------------------
__global__ __launch_bounds__(32)
void qgemm(const float* __restrict__ probs, const float* __restrict__ Wp,
           const float* __restrict__ bias, float* __restrict__ out, int nb)
{
    const int lane = threadIdx.x;
    const int half = lane >> 4;
    const int l    = lane & 15;

    int rowA = blockIdx.x * 16 + l;
    if (rowA >= nb) rowA = nb - 1;              // safe clamp for loads
    const float* aRow = probs + (size_t)rowA * SSTATES;

    v8f acc = {};
    for (int k = 0; k < SSTATES; k += 4) {
        const int ka = k + half * 2;
        v2f a, bv;
        a.x  = aRow[ka + 0];
        a.y  = aRow[ka + 1];
        bv.x = Wp[(size_t)(ka + 0) * 16 + l];
        bv.y = Wp[(size_t)(ka + 1) * 16 + l];
        acc = __builtin_amdgcn_wmma_f32_16x16x4_f32(
                  false, a, false, bv, (short)0, acc, false, false);
    }

    const float bo = (l < OUTF) ? bias[l] : 0.0f;
    #pragma unroll
    for (int i = 0; i < 8; ++i) {
        const int row = blockIdx.x * 16 + half * 8 + i;
        if (row < nb && l < OUTF)
            out[(size_t)row * OUTF + l] = acc[i] + bo;
    }
}

// ---------------------------------------------------------------------------
// Host launcher
// ---------------------------------------------------------------------------
extern "C" void kernel_launch(void* const* d_in, const int* in_sizes, int n_in,
                              void* d_out, int out_size, void* d_ws, size_t ws_size,
                              hipStream_t stream)
{
    const float* x      = (const float*)d_in[0];
    const float* thetas = (const float*)d_in[1];
    const float* W      = (const float*)d_in[2];
    const float* bias   = (const float*)d_in[3];
    float*       out    = (float*)d_out;

    const int nb = in_sizes[0] / NXF;   // batch (512)

    // workspace layout (256-byte aligned sections)
    const size_t offIdx = 0;                         // 4368 * 4  = 17472
    const size_t offInv = 17664;                     // 4368 * 4  = 17472
    const size_t offWp  = 35328;                     // 4368 * 16 * 4 = 279552
    const size_t offP   = 35328 + 279552;            // 314880
    const size_t need   = offP + (size_t)nb * SSTATES * sizeof(float);

    uint32_t* idxPack = (uint32_t*)((char*)d_ws + offIdx);
    float*    invnorm = (float*)((char*)d_ws + offInv);

    const bool wmmaPath = (ws_size >= need);
    float* Wp    = wmmaPath ? (float*)((char*)d_ws + offWp) : nullptr;
    float* probs = wmmaPath ? (float*)((char*)d_ws + offP)  : nullptr;

    qinit<<<(SSTATES + 255) / 256, 256, 0, stream>>>(idxPack, invnorm, W, Wp);

    if (wmmaPath) {
        qmain<false><<<nb, 256, 0, stream>>>(x, thetas, idxPack, invnorm,
                                             probs, nullptr, nullptr, nullptr);
        qgemm<<<(nb + 15) / 16, 32, 0, stream>>>(probs, Wp, bias, out, nb);
    } else {
        qmain<true><<<nb, 256, 0, stream>>>(x, thetas, idxPack, invnorm,
                                            nullptr, W, bias, out);
    }
}